// GraphSAGE_59390807769650
// MI455X (gfx1250) — compile-verified
//
#include <hip/hip_runtime.h>

typedef __attribute__((ext_vector_type(2))) float v2f;
typedef __attribute__((ext_vector_type(8))) float v8f;

#define IN_F  128
#define HID_F 128
#define OUT_F 64

// ---------------- utility kernels ----------------

__global__ __launch_bounds__(256) void zero_kernel(float* __restrict__ p, long long n) {
    long long i = (long long)blockIdx.x * 256 + threadIdx.x;
    if (i < n) p[i] = 0.0f;
}

__global__ __launch_bounds__(256) void degree_kernel(const int* __restrict__ dst,
                                                     float* __restrict__ deg, int E) {
    int e = blockIdx.x * 256 + threadIdx.x;
    if (e < E) atomicAdd(&deg[dst[e]], 1.0f);
}

__global__ __launch_bounds__(256) void invdeg_kernel(float* __restrict__ deg, int n) {
    int i = blockIdx.x * 256 + threadIdx.x;
    if (i < n) deg[i] = 1.0f / fmaxf(deg[i], 1.0f);
}

// Scatter-add of a 128-wide feature row along each edge: accum[dst] += feat[src].
// One thread per (edge, dim). Accumulator (~25MB) is L2-resident on MI455X (192MB L2).
__global__ __launch_bounds__(256) void scatter_kernel(const float* __restrict__ feat,
                                                      const int* __restrict__ src,
                                                      const int* __restrict__ dst,
                                                      float* __restrict__ accum, int E) {
    long long tid = (long long)blockIdx.x * 256 + threadIdx.x;
    int e = (int)(tid >> 7);          // 128 dims per edge
    int d = (int)(tid & 127);
    if (e < E) {
        int s = src[e], t = dst[e];
        atomicAdd(&accum[(long long)t * 128 + d], feat[(long long)s * 128 + d]);
    }
}

// ---------------- fused SAGE layer GEMM via V_WMMA_F32_16X16X4_F32 ----------------
// out[M x N] = act( Xs @ Ws + (Xa * invdeg_row) @ Wn + bias )
// One wave per 16x16 output tile; NCOLT waves per block cover all N/16 column tiles
// of one row tile (A fragments are shared through L0/L2).
template <int NCOLT>
__global__ __launch_bounds__(32 * NCOLT) void sage_gemm_wmma(
        const float* __restrict__ Xs, const float* __restrict__ Xa,
        const float* __restrict__ invdeg,
        const float* __restrict__ Ws, const float* __restrict__ Wn,
        const float* __restrict__ bias, float* __restrict__ out,
        int M, int K, int do_relu) {
    const int N = NCOLT * 16;
    const int lane    = threadIdx.x;              // 0..31, wave32
    const int rowBase = blockIdx.x * 16;
    const int colBase = threadIdx.y * 16;

    // A-matrix 16x4 f32 layout: lane&15 = M row; lanes>=16 hold K=2,3 (VGPR0=K0/K2, VGPR1=K1/K3)
    int r = rowBase + (lane & 15);
    if (r >= M) r = M - 1;                        // tail-safe loads (M here is a multiple of 16)
    const int kOff = (lane < 16) ? 0 : 2;
    // B-matrix 4x16 f32 layout: lane&15 = N col; lanes>=16 hold K=2,3, one K per VGPR
    const int c = colBase + (lane & 15);

    const float invd = invdeg[r];
    const float* xsRow = Xs + (long long)r * K + kOff;
    const float* xaRow = Xa + (long long)r * K + kOff;

    v8f acc = {};
    for (int kk = 0; kk < K; kk += 4) {
        v2f a_s = *reinterpret_cast<const v2f*>(xsRow + kk);          // K = kk+kOff, kk+kOff+1
        v2f a_raw = *reinterpret_cast<const v2f*>(xaRow + kk);
        v2f a_n = { a_raw.x * invd, a_raw.y * invd };                  // fused neighbor-mean
        const int kb = kk + kOff;
        v2f b_s = { Ws[(long long)kb * N + c], Ws[(long long)(kb + 1) * N + c] };
        v2f b_n = { Wn[(long long)kb * N + c], Wn[(long long)(kb + 1) * N + c] };
        acc = __builtin_amdgcn_wmma_f32_16x16x4_f32(false, a_s, false, b_s,
                                                    (short)0, acc, false, false);
        acc = __builtin_amdgcn_wmma_f32_16x16x4_f32(false, a_n, false, b_n,
                                                    (short)0, acc, false, false);
    }

    // C/D 16x16 f32 layout: VGPR j -> M = rowBase + j (+8 for lanes 16..31), N = colBase + (lane&15)
    const float bv = bias[c];
    const int mBase = rowBase + ((lane < 16) ? 0 : 8);
#pragma unroll
    for (int j = 0; j < 8; ++j) {
        int m = mBase + j;
        if (m < M) {
            float v = acc[j] + bv;
            if (do_relu) v = fmaxf(v, 0.0f);
            out[(long long)m * N + c] = v;
        }
    }
}

// ---------------- per-graph mean readout ----------------

__global__ __launch_bounds__(256) void gcount_kernel(const int* __restrict__ gid,
                                                     float* __restrict__ gcnt, int n) {
    int i = blockIdx.x * 256 + threadIdx.x;
    if (i < n) atomicAdd(&gcnt[gid[i]], 1.0f);
}

__global__ __launch_bounds__(256) void readout_kernel(const float* __restrict__ h2,
                                                      const int* __restrict__ gid,
                                                      float* __restrict__ gsum, int Nn) {
    long long tid = (long long)blockIdx.x * 256 + threadIdx.x;
    int node = (int)(tid >> 6);        // OUT_F = 64
    int f    = (int)(tid & 63);
    if (node < Nn)
        atomicAdd(&gsum[(long long)gid[node] * OUT_F + f], h2[(long long)node * OUT_F + f]);
}

__global__ __launch_bounds__(256) void finalize_kernel(const float* __restrict__ gsum,
                                                       const float* __restrict__ gcnt,
                                                       float* __restrict__ out, int total) {
    int i = blockIdx.x * 256 + threadIdx.x;
    if (i < total) {
        int g = i >> 6;                // OUT_F = 64
        out[i] = gsum[i] / fmaxf(gcnt[g], 1.0f);
    }
}

// ---------------- launch ----------------

static inline int cdiv64(long long a, long long b) { return (int)((a + b - 1) / b); }

extern "C" void kernel_launch(void* const* d_in, const int* in_sizes, int n_in,
                              void* d_out, int out_size, void* d_ws, size_t ws_size,
                              hipStream_t stream) {
    const float* in_feat = (const float*)d_in[0];
    const int*   src     = (const int*)d_in[1];
    const int*   dst     = (const int*)d_in[2];
    const int*   gid     = (const int*)d_in[3];
    // d_in[4] = num_graphs (device scalar) — G derived from out_size instead
    const float* W1s = (const float*)d_in[5];
    const float* W1n = (const float*)d_in[6];
    const float* b1  = (const float*)d_in[7];
    const float* W2s = (const float*)d_in[8];
    const float* W2n = (const float*)d_in[9];
    const float* b2  = (const float*)d_in[10];
    float* out = (float*)d_out;

    const int Nn = in_sizes[0] / IN_F;   // 50000
    const int E  = in_sizes[1];          // 600000
    const int G  = out_size / OUT_F;     // 64

    // workspace carve-up (floats)
    float* ws    = (float*)d_ws;
    float* accum = ws;                                   // Nn*128
    float* deg   = accum + (size_t)Nn * 128;             // Nn (becomes invdeg in place)
    float* h1    = deg + Nn;                             // Nn*128
    float* h2    = h1 + (size_t)Nn * 128;                // Nn*64
    float* gsum  = h2 + (size_t)Nn * OUT_F;              // G*64
    float* gcnt  = gsum + (size_t)G * OUT_F;             // G

    const int mTiles = cdiv64(Nn, 16);

    // ---- layer 1: mean-aggregate raw features, fused GEMM + ReLU ----
    zero_kernel<<<cdiv64((long long)Nn * 128 + Nn, 256), 256, 0, stream>>>(
        accum, (long long)Nn * 128 + Nn);                // accum + deg contiguous
    degree_kernel<<<cdiv64(E, 256), 256, 0, stream>>>(dst, deg, E);
    scatter_kernel<<<cdiv64((long long)E * 128, 256), 256, 0, stream>>>(in_feat, src, dst, accum, E);
    invdeg_kernel<<<cdiv64(Nn, 256), 256, 0, stream>>>(deg, Nn);
    sage_gemm_wmma<8><<<mTiles, dim3(32, 8), 0, stream>>>(
        in_feat, accum, deg, W1s, W1n, b1, h1, Nn, IN_F, 1);

    // ---- layer 2: mean-aggregate h1, fused GEMM (no ReLU) ----
    zero_kernel<<<cdiv64((long long)Nn * 128, 256), 256, 0, stream>>>(accum, (long long)Nn * 128);
    scatter_kernel<<<cdiv64((long long)E * 128, 256), 256, 0, stream>>>(h1, src, dst, accum, E);
    sage_gemm_wmma<4><<<mTiles, dim3(32, 4), 0, stream>>>(
        h1, accum, deg, W2s, W2n, b2, h2, Nn, HID_F, 0);

    // ---- per-graph mean readout ----
    zero_kernel<<<cdiv64((long long)G * OUT_F + G, 256), 256, 0, stream>>>(
        gsum, (long long)G * OUT_F + G);                 // gsum + gcnt contiguous
    gcount_kernel<<<cdiv64(Nn, 256), 256, 0, stream>>>(gid, gcnt, Nn);
    readout_kernel<<<cdiv64((long long)Nn * OUT_F, 256), 256, 0, stream>>>(h2, gid, gsum, Nn);
    finalize_kernel<<<cdiv64((long long)G * OUT_F, 256), 256, 0, stream>>>(gsum, gcnt, out, G * OUT_F);
}